// EdgeGATLayer_57140244906497
// MI455X (gfx1250) — compile-verified
//
#include <hip/hip_runtime.h>
#include <hip/hip_bf16.h>

typedef float v2f __attribute__((ext_vector_type(2)));
typedef float v8f __attribute__((ext_vector_type(8)));

#define DE 128   // edge embed size (== node embed size here)

// Branch-free tanh: (e^2x - 1)/(e^2x + 1), clamped so exp never overflows.
__device__ __forceinline__ float egat_tanh(float x) {
    x = fminf(fmaxf(x, -15.0f), 15.0f);
    float t = __expf(2.0f * x);
    return (t - 1.0f) / (t + 1.0f);
}

// ---------------------------------------------------------------------------
// Kernel 1: zero the segment-sum + count accumulators in workspace (float4)
// ---------------------------------------------------------------------------
__global__ void egat_zero_f4(float4* __restrict__ p, long n4) {
    long i = (long)blockIdx.x * blockDim.x + threadIdx.x;
    long stride = (long)gridDim.x * blockDim.x;
    float4 z = {0.f, 0.f, 0.f, 0.f};
    for (long j = i; j < n4; j += stride) p[j] = z;
}

// ---------------------------------------------------------------------------
// Kernel 2: edge scatter-add for one relation.
// One wave per edge, each lane handles 4 consecutive dims (float4 load,
// 4x global_atomic_add_f32 no-return). Lane 0 bumps the count.
// ---------------------------------------------------------------------------
__global__ __launch_bounds__(256)
void egat_scatter(const float* __restrict__ E, const int* __restrict__ dst,
                  float* __restrict__ sums, float* __restrict__ cnt, int n_edges) {
    int gid  = blockIdx.x * blockDim.x + threadIdx.x;
    int edge = gid >> 5;
    int lane = gid & 31;
    if (edge >= n_edges) return;
    int d = dst[edge];
    float4 v = ((const float4*)(E + (long)edge * DE))[lane];
    float* s = sums + (long)d * DE + lane * 4;
    unsafeAtomicAdd(s + 0, v.x);
    unsafeAtomicAdd(s + 1, v.y);
    unsafeAtomicAdd(s + 2, v.z);
    unsafeAtomicAdd(s + 3, v.w);
    if (lane == 0) unsafeAtomicAdd(cnt + d, 1.0f);
}

// ---------------------------------------------------------------------------
// Kernel 3: Y = sums / max(count,1)   (in place, mean reducer; empty -> 0)
// ---------------------------------------------------------------------------
__global__ __launch_bounds__(256)
void egat_mean_div(float* __restrict__ Y, const float* __restrict__ cnt, long rows) {
    long gid = (long)blockIdx.x * blockDim.x + threadIdx.x;
    long row = gid >> 5;
    int  q   = gid & 31;
    if (row >= rows) return;
    float inv = 1.0f / fmaxf(cnt[row], 1.0f);
    float4* y4 = (float4*)(Y + row * (long)DE);
    float4 v = y4[q];
    v.x *= inv; v.y *= inv; v.z *= inv; v.w *= inv;
    y4[q] = v;
}

// ---------------------------------------------------------------------------
// Shared helper: stage W [128x128] into LDS K-pair interleaved:
//   sW2[(k>>1)*DE + col] = { W[k][col], W[k+1][col] }   (k even)
// so a lane's B operand (two K values, same col) is ONE aligned ds_load_b64
// landing in an adjacent (even) VGPR pair -- exactly what WMMA wants.
// ---------------------------------------------------------------------------
__device__ __forceinline__ void egat_stage_W(float2* sW2, const float* __restrict__ W) {
    for (int idx = threadIdx.x; idx < DE * DE; idx += blockDim.x) {
        int k = idx / DE, col = idx % DE;
        ((float*)&sW2[(k >> 1) * DE + col])[k & 1] = W[idx];
    }
}

// ---------------------------------------------------------------------------
// Kernel 4: e[row] = (tanh(Y[row,:] @ W1) @ W2)   via v_wmma_f32_16x16x4_f32.
// One wave owns a 16-row tile; 8 column-tile accumulators (full 16x128 T tile
// lives in VGPRs, never hits memory). W1 staged in LDS (pair-interleaved).
//
// f32 WMMA operand layout (ISA 7.12.2):
//   A 16x4 : lane l -> row = l%16 ; VGPR0/1 hold K = 2*(l/16) + {0,1}
//   B 4x16 : lane l -> col = l%16 ; VGPR0/1 hold K = 2*(l/16) + {0,1}
//   C/D    : lane l -> col = l%16 ; VGPR i holds row = i + 8*(l/16)
// ---------------------------------------------------------------------------
__global__ __launch_bounds__(256)
void egat_fc1_tanh_fc2(const float* __restrict__ Y, const float* __restrict__ W1,
                       const float* __restrict__ W2, float* __restrict__ evec,
                       int rows) {
    __shared__ float2 sW2[(DE / 2) * DE];   // 64 KB
    egat_stage_W(sW2, W1);
    __syncthreads();

    int waveId  = threadIdx.x >> 5;
    int lane    = threadIdx.x & 31;
    int half    = lane >> 4;       // 0 or 1
    int l16     = lane & 15;
    int rowBase = (blockIdx.x * (blockDim.x >> 5) + waveId) * 16;
    if (rowBase >= rows) return;

    int arow = rowBase + l16;
    if (arow > rows - 1) arow = rows - 1;          // clamp (padded rows unused)
    const float2* Arow2 = (const float2*)(Y + (long)arow * DE);

    v8f acc[8];
#pragma unroll
    for (int t = 0; t < 8; ++t) acc[t] = (v8f){0,0,0,0,0,0,0,0};

    for (int kk = 0; kk < DE; kk += 4) {
        int kp = (kk >> 1) + half;                 // K-pair index for this lane
        float2 a2 = Arow2[kp];                     // global_load_b64
        v2f a; a.x = a2.x; a.y = a2.y;
#pragma unroll
        for (int t = 0; t < 8; ++t) {
            float2 b2 = sW2[kp * DE + t * 16 + l16];   // ds_load_b64
            v2f b; b.x = b2.x; b.y = b2.y;
            acc[t] = __builtin_amdgcn_wmma_f32_16x16x4_f32(
                false, a, false, b, (short)0, acc[t], false, false);
        }
    }

    // e[row] = sum_col tanh(T[row,col]) * W2[col]; reduce 16 lanes per half-wave
    float er[8];
#pragma unroll
    for (int i = 0; i < 8; ++i) er[i] = 0.f;
#pragma unroll
    for (int t = 0; t < 8; ++t) {
        float w2c = W2[t * 16 + l16];
#pragma unroll
        for (int i = 0; i < 8; ++i) er[i] += egat_tanh(acc[t][i]) * w2c;
    }
#pragma unroll
    for (int i = 0; i < 8; ++i) {
        er[i] += __shfl_xor(er[i], 1, 16);
        er[i] += __shfl_xor(er[i], 2, 16);
        er[i] += __shfl_xor(er[i], 4, 16);
        er[i] += __shfl_xor(er[i], 8, 16);
    }
    if (l16 == 0) {
#pragma unroll
        for (int i = 0; i < 8; ++i) {
            int r = rowBase + i + 8 * half;
            if (r < rows) evec[r] = er[i];
        }
    }
}

// ---------------------------------------------------------------------------
// Kernel 5: 3-way softmax over relations + weighted combine -> OY [N x 128]
// ---------------------------------------------------------------------------
__global__ __launch_bounds__(256)
void egat_softmax_combine(const float* __restrict__ Y, const float* __restrict__ evec,
                          float* __restrict__ OY, int N) {
    long gid = (long)blockIdx.x * blockDim.x + threadIdx.x;
    long n = gid >> 5;
    int  q = gid & 31;
    if (n >= N) return;
    float e0 = evec[n], e1 = evec[N + n], e2 = evec[2L * N + n];
    float m  = fmaxf(e0, fmaxf(e1, e2));
    float a0 = __expf(e0 - m), a1 = __expf(e1 - m), a2 = __expf(e2 - m);
    float inv = 1.0f / (a0 + a1 + a2);
    a0 *= inv; a1 *= inv; a2 *= inv;
    float4 v0 = ((const float4*)(Y + n * (long)DE))[q];
    float4 v1 = ((const float4*)(Y + ((long)N + n) * DE))[q];
    float4 v2 = ((const float4*)(Y + (2L * N + n) * DE))[q];
    float4 o;
    o.x = a0 * v0.x + a1 * v1.x + a2 * v2.x;
    o.y = a0 * v0.y + a1 * v1.y + a2 * v2.y;
    o.z = a0 * v0.z + a1 * v1.z + a2 * v2.z;
    o.w = a0 * v0.w + a1 * v1.w + a2 * v2.w;
    ((float4*)(OY + n * (long)DE))[q] = o;
}

// ---------------------------------------------------------------------------
// Kernel 6: out = OY @ W3   (N x 128) @ (128 x 128), same WMMA structure
// ---------------------------------------------------------------------------
__global__ __launch_bounds__(256)
void egat_gemm_out(const float* __restrict__ X, const float* __restrict__ W,
                   float* __restrict__ out, int rows) {
    __shared__ float2 sW2[(DE / 2) * DE];   // 64 KB
    egat_stage_W(sW2, W);
    __syncthreads();

    int waveId  = threadIdx.x >> 5;
    int lane    = threadIdx.x & 31;
    int half    = lane >> 4;
    int l16     = lane & 15;
    int rowBase = (blockIdx.x * (blockDim.x >> 5) + waveId) * 16;
    if (rowBase >= rows) return;

    int arow = rowBase + l16;
    if (arow > rows - 1) arow = rows - 1;
    const float2* Arow2 = (const float2*)(X + (long)arow * DE);

    v8f acc[8];
#pragma unroll
    for (int t = 0; t < 8; ++t) acc[t] = (v8f){0,0,0,0,0,0,0,0};

    for (int kk = 0; kk < DE; kk += 4) {
        int kp = (kk >> 1) + half;
        float2 a2 = Arow2[kp];
        v2f a; a.x = a2.x; a.y = a2.y;
#pragma unroll
        for (int t = 0; t < 8; ++t) {
            float2 b2 = sW2[kp * DE + t * 16 + l16];
            v2f b; b.x = b2.x; b.y = b2.y;
            acc[t] = __builtin_amdgcn_wmma_f32_16x16x4_f32(
                false, a, false, b, (short)0, acc[t], false, false);
        }
    }
#pragma unroll
    for (int t = 0; t < 8; ++t) {
        int col = t * 16 + l16;
#pragma unroll
        for (int i = 0; i < 8; ++i) {
            int r = rowBase + i + 8 * half;
            if (r < rows) out[(long)r * DE + col] = acc[t][i];
        }
    }
}

// ---------------------------------------------------------------------------
// Host launcher
// inputs: 0:E_C 1:E_D 2:E_M 3:dst_C 4:dst_D 5:dst_M 6:num_nodes 7:W1 8:W2 9:W3
// ---------------------------------------------------------------------------
extern "C" void kernel_launch(void* const* d_in, const int* in_sizes, int n_in,
                              void* d_out, int out_size, void* d_ws, size_t ws_size,
                              hipStream_t stream) {
    const float* E[3]   = {(const float*)d_in[0], (const float*)d_in[1], (const float*)d_in[2]};
    const int*   dst[3] = {(const int*)d_in[3],   (const int*)d_in[4],   (const int*)d_in[5]};
    const float* W1 = (const float*)d_in[7];
    const float* W2 = (const float*)d_in[8];
    const float* W3 = (const float*)d_in[9];
    float* out = (float*)d_out;

    const int  n_edges = in_sizes[0] / DE;
    const int  N       = out_size / DE;       // DN == 128
    const long rows3   = 3L * N;

    // workspace layout (floats): [Y sums 3N*128][counts 3N][e 3N][OY N*128]
    float* ws   = (float*)d_ws;
    float* Yall = ws;
    float* cnt  = Yall + rows3 * DE;
    float* evec = cnt + rows3;
    float* OY   = evec + rows3;

    // 1) zero sums + counts (contiguous, multiple of 4 floats)
    {
        long n4 = (rows3 * DE + rows3) / 4;
        egat_zero_f4<<<4096, 256, 0, stream>>>((float4*)Yall, n4);
    }
    // 2) scatter-add per relation (one wave per edge)
    {
        int threads = 256;
        int blocks  = (int)(((long)n_edges * 32 + threads - 1) / threads);
        for (int r = 0; r < 3; ++r)
            egat_scatter<<<blocks, threads, 0, stream>>>(
                E[r], dst[r], Yall + (long)r * N * DE, cnt + (long)r * N, n_edges);
    }
    // 3) mean divide
    {
        long threads_total = rows3 * 32;
        int blocks = (int)((threads_total + 255) / 256);
        egat_mean_div<<<blocks, 256, 0, stream>>>(Yall, cnt, rows3);
    }
    // 4) e = tanh(Y @ W1) @ W2   (WMMA; 8 waves/block, one 16-row tile each)
    {
        int rowTiles = (int)((rows3 + 15) / 16);
        int blocks   = (rowTiles + 7) / 8;
        egat_fc1_tanh_fc2<<<blocks, 256, 0, stream>>>(Yall, W1, W2, evec, (int)rows3);
    }
    // 5) softmax over relations + combine -> OY
    {
        long threads_total = (long)N * 32;
        int blocks = (int)((threads_total + 255) / 256);
        egat_softmax_combine<<<blocks, 256, 0, stream>>>(Yall, evec, OY, N);
    }
    // 6) out = OY @ W3 (WMMA)
    {
        int rowTiles = (N + 15) / 16;
        int blocks   = (rowTiles + 7) / 8;
        egat_gemm_out<<<blocks, 256, 0, stream>>>(OY, W3, out, N);
    }
}